// AngularAttention_20452634263753
// MI455X (gfx1250) — compile-verified
//
#include <hip/hip_runtime.h>
#include <hip/hip_bf16.h>

// ---------------------------------------------------------------------------
// AngularAttention for MI455X (gfx1250), wave32 + V_WMMA_F32_16X16X4_F32.
//
// Shapes: x[4][64][25][64][64] f32, W[192][64] f32, out[4][64][25][64][64].
// Strategy (bandwidth-bound, ~15us ideal @ 23.3 TB/s):
//   K0: Cm = Wq^T Wk, A = Wq^T Wq, B = Wk^T Wk (64x64 each) + zero stats.
//   K1: per batch, fp32-WMMA SYRK of X (1600x4096 contiguous) in 64x64 tiles;
//       fuse contraction with Cm/A/B in registers -> LDS atomics -> global
//       atomics producing S[25][25], |q|^2, |k|^2.  No q/k ever stored.
//   K2: att = softmax(S / (max(|q|,eps) max(|k|,eps))).
//   K3: out = Wv (att x): WMMA stage1 V=Wv*X into LDS, WMMA stage2 att-mix.
// ---------------------------------------------------------------------------

typedef __attribute__((ext_vector_type(2))) float v2f;
typedef __attribute__((ext_vector_type(8))) float v8f;

#define CH        64
#define NANG      25
#define HW        4096                 // 64*64 spatial positions
#define ROWS      1600                 // CH*NANG, rows of X (contiguous 4096-f32 rows)
#define XBSTRIDE  (CH * NANG * HW)     // per-batch x stride (floats)

// workspace layout (floats)
#define WS_CM   0                      // 4096
#define WS_A    4096                   // 4096
#define WS_B    8192                   // 4096
#define WS_S    12288                  // 4*625
#define WS_NQ   14788                  // 4*25
#define WS_NK   14888                  // 4*25
#define WS_ATT  14988                  // 4*625

static __device__ __forceinline__ v8f wmma_f32(v2f a, v2f b, v8f c) {
    // D(16x16,f32) = A(16x4,f32) * B(4x16,f32) + C
    return __builtin_amdgcn_wmma_f32_16x16x4_f32(false, a, false, b, (short)0, c,
                                                 false, false);
}

// ---------------------------------------------------------------------------
// K0: zero the stat accumulators; build the three 64x64 weight Grams.
// ---------------------------------------------------------------------------
__global__ void k_prep(const float* __restrict__ Wm, float* __restrict__ ws) {
    int t = threadIdx.x;
    for (int i = t; i < 2700; i += 256) ws[WS_S + i] = 0.0f;  // S, nq, nk contiguous
    for (int idx = t; idx < 3 * 4096; idx += 256) {
        int which = idx >> 12;            // 0: Wq^T Wk, 1: Wq^T Wq, 2: Wk^T Wk
        int e = idx & 4095, c = e >> 6, c2 = e & 63;
        int base1 = (which == 2) ? 64 : 0;
        int base2 = (which == 0 || which == 2) ? 64 : 0;
        float s = 0.0f;
        for (int d = 0; d < 64; ++d)
            s = fmaf(Wm[(base1 + d) * 64 + c], Wm[(base2 + d) * 64 + c2], s);
        ws[idx] = s;
    }
}

// ---------------------------------------------------------------------------
// K1: fused Gram + contraction.  Grid (25,25,4), block 128 (4 waves).
// Each workgroup: 64x64 tile of G_b = X_b X_b^T, K=4096 staged through LDS.
// ---------------------------------------------------------------------------
__global__ void k_gram_stats(const float* __restrict__ x, float* __restrict__ ws) {
    __shared__ float Xa[64 * 36];      // row stride 36 -> conflict-free frag reads
    __shared__ float Xb[64 * 36];
    __shared__ float Sl[625];
    __shared__ float Nql[25];
    __shared__ float Nkl[25];

    const int t = threadIdx.x, lane = t & 31, w = t >> 5;
    const int b = blockIdx.z;
    const int r0 = blockIdx.x * 64, c0 = blockIdx.y * 64;
    const int wi = w >> 1, wj = w & 1;            // 32x32 quadrant per wave
    const int l15 = lane & 15, hi = lane >> 4;

    for (int i = t; i < 625; i += 128) Sl[i] = 0.0f;
    if (t < 25) { Nql[t] = 0.0f; Nkl[t] = 0.0f; }
    __syncthreads();

    const float* xb = x + (size_t)b * XBSTRIDE;
    const v8f vzero = {0.f, 0.f, 0.f, 0.f, 0.f, 0.f, 0.f, 0.f};
    v8f acc[2][2];
    acc[0][0] = vzero; acc[0][1] = vzero; acc[1][0] = vzero; acc[1][1] = vzero;

    for (int k0 = 0; k0 < HW; k0 += 32) {
        // stage 64 rows x 32 k-cols of both row-block and col-block (coalesced f4)
        for (int q = 0; q < 4; ++q) {
            int slot = t + 128 * q;                       // 0..511
            int row = slot >> 3, seg = slot & 7;
            float4 va = *(const float4*)(xb + (size_t)(r0 + row) * HW + k0 + seg * 4);
            float4 vb = *(const float4*)(xb + (size_t)(c0 + row) * HW + k0 + seg * 4);
            *(float4*)&Xa[row * 36 + seg * 4] = va;
            *(float4*)&Xb[row * 36 + seg * 4] = vb;
        }
        __syncthreads();

        for (int kk = 0; kk < 8; ++kk) {
            int kb = kk * 4 + hi * 2;
            v2f aM[2], bN[2];
            for (int m = 0; m < 2; ++m) {
                int ra = wi * 32 + m * 16 + l15;          // A: M=row, K striped
                aM[m].x = Xa[ra * 36 + kb];
                aM[m].y = Xa[ra * 36 + kb + 1];
                int rb = wj * 32 + m * 16 + l15;          // B[K][N] = Xb[N][K]
                bN[m].x = Xb[rb * 36 + kb];
                bN[m].y = Xb[rb * 36 + kb + 1];
            }
            acc[0][0] = wmma_f32(aM[0], bN[0], acc[0][0]);
            acc[0][1] = wmma_f32(aM[0], bN[1], acc[0][1]);
            acc[1][0] = wmma_f32(aM[1], bN[0], acc[1][0]);
            acc[1][1] = wmma_f32(aM[1], bN[1], acc[1][1]);
        }
        __syncthreads();
    }

    // contract G-tile with Cm / A / B ;  row r = c*25 + n (x layout [c][n][p])
    const float* Cm = ws + WS_CM;
    const float* Am = ws + WS_A;
    const float* Bm = ws + WS_B;
    for (int mi = 0; mi < 2; ++mi)
        for (int nj = 0; nj < 2; ++nj) {
            v8f g = acc[mi][nj];
            int rbase = r0 + wi * 32 + mi * 16 + hi * 8;
            int rr = c0 + wj * 32 + nj * 16 + l15;
            int cB = rr / 25, m = rr - cB * 25;
            for (int i = 0; i < 8; ++i) {
                float gv = g[i];
                int r = rbase + i;
                int cA = r / 25, n = r - cA * 25;
                atomicAdd(&Sl[n * 25 + m], Cm[cA * 64 + cB] * gv);
                if (n == m) {
                    atomicAdd(&Nql[n], Am[cA * 64 + cB] * gv);
                    atomicAdd(&Nkl[n], Bm[cA * 64 + cB] * gv);
                }
            }
        }
    __syncthreads();

    float* Sg = ws + WS_S + b * 625;
    for (int i = t; i < 625; i += 128) atomicAdd(&Sg[i], Sl[i]);
    if (t < 25) {
        atomicAdd(&ws[WS_NQ + b * 25 + t], Nql[t]);
        atomicAdd(&ws[WS_NK + b * 25 + t], Nkl[t]);
    }
}

// ---------------------------------------------------------------------------
// K2: normalize + softmax over m.  Grid 4, block 32.
// ---------------------------------------------------------------------------
__global__ void k_softmax(float* __restrict__ ws) {
    int b = blockIdx.x, n = threadIdx.x;
    if (n >= 25) return;
    const float* S = ws + WS_S + b * 625;
    const float* nq = ws + WS_NQ + b * 25;
    const float* nk = ws + WS_NK + b * 25;
    float* att = ws + WS_ATT + b * 625;
    float qn = fmaxf(sqrtf(fmaxf(nq[n], 0.0f)), 1e-12f);
    float row[25], mx = -1e30f;
    for (int m = 0; m < 25; ++m) {
        float kn = fmaxf(sqrtf(fmaxf(nk[m], 0.0f)), 1e-12f);
        row[m] = S[n * 25 + m] / (qn * kn);
        mx = fmaxf(mx, row[m]);
    }
    float sum = 0.0f;
    for (int m = 0; m < 25; ++m) { row[m] = __expf(row[m] - mx); sum += row[m]; }
    float inv = 1.0f / sum;
    for (int m = 0; m < 25; ++m) att[n * 25 + m] = row[m] * inv;
}

// ---------------------------------------------------------------------------
// K3: out = Wv (att x).  Grid (512,1,4) over 8-position chunks, block 256.
// Stage1 (WMMA): V[m][d][p] = Wv * X  (x read exactly once, B-frags reused
// across 4 d-tiles in registers).  Stage2 (WMMA): out[n] = att[n][m] V[m].
// ---------------------------------------------------------------------------
__global__ void k_out(const float* __restrict__ x, const float* __restrict__ Wm,
                      const float* __restrict__ ws, float* __restrict__ out) {
    __shared__ float Vl[25 * 64 * 8];      // [m][d][p]   51200 B
    __shared__ float attp[32 * 28];        // padded att   3584 B

    const int t = threadIdx.x, lane = t & 31, w = t >> 5;
    const int b = blockIdx.z, p0 = blockIdx.x * 8;
    const int l15 = lane & 15, hi = lane >> 4;
    const float* xb = x + (size_t)b * XBSTRIDE;
    const float* att = ws + WS_ATT + b * 625;
    const v8f vzero = {0.f, 0.f, 0.f, 0.f, 0.f, 0.f, 0.f, 0.f};

    for (int i = t; i < 32 * 28; i += 256) {
        int r = i / 28, c = i - r * 28;
        attp[i] = (r < 25 && c < 25) ? att[r * 25 + c] : 0.0f;
    }
    __syncthreads();

    // ---- stage 1: V = Wv * X ------------------------------------------------
    {
        const int mloc = l15 >> 3, pl = l15 & 7;       // B cols = (m pair, 8 pos)
        for (int mp = w; mp < 13; mp += 8) {
            int m0 = mp * 2;
            int mB = min(m0 + mloc, 24);               // dup last m; store-masked
            v2f bfr[16];
            for (int kk = 0; kk < 16; ++kk) {          // K = channel c
                int c = kk * 4 + hi * 2;
                const float* px = xb + ((size_t)c * 25 + mB) * HW + p0 + pl;
                bfr[kk].x = px[0];
                bfr[kk].y = px[(size_t)25 * HW];       // channel c+1
            }
            for (int dt = 0; dt < 4; ++dt) {
                v8f acc = vzero;
                for (int kk = 0; kk < 16; ++kk) {
                    int cc = kk * 4 + hi * 2;
                    int drow = 128 + dt * 16 + l15;    // Wv = W rows 128..191
                    v2f a;
                    a.x = Wm[drow * 64 + cc];
                    a.y = Wm[drow * 64 + cc + 1];
                    acc = wmma_f32(a, bfr[kk], acc);
                }
                if (m0 + mloc < 25) {
                    int mdst = m0 + mloc;
                    for (int i = 0; i < 8; ++i) {
                        int d = dt * 16 + i + hi * 8;
                        Vl[mdst * 512 + d * 8 + pl] = acc[i];
                    }
                }
            }
        }
    }
    __syncthreads();

    // ---- stage 2: out[n] = att[n][m] * V[m] --------------------------------
    {
        const int dloc = l15 >> 3, pl = l15 & 7;
        for (int ii = 0; ii < 8; ++ii) {
            int T = w * 8 + ii;                        // 64 C-tiles total
            int ntile = T >> 5, j = T & 31;
            int d = j * 2 + dloc;
            v8f acc = vzero;
            for (int kk = 0; kk < 7; ++kk) {           // K = m padded 25 -> 28
                int K = kk * 4 + hi * 2;
                v2f a, bb;
                a.x = attp[(ntile * 16 + l15) * 28 + K];
                a.y = attp[(ntile * 16 + l15) * 28 + K + 1];
                int m1 = min(K, 24), m2 = min(K + 1, 24);   // att pad is 0 -> safe
                bb.x = Vl[m1 * 512 + d * 8 + pl];
                bb.y = Vl[m2 * 512 + d * 8 + pl];
                acc = wmma_f32(a, bb, acc);
            }
            for (int i = 0; i < 8; ++i) {
                int n = ntile * 16 + i + hi * 8;
                if (n < 25)
                    out[(((size_t)b * 64 + d) * 25 + n) * HW + p0 + pl] = acc[i];
            }
        }
    }
}

// ---------------------------------------------------------------------------
extern "C" void kernel_launch(void* const* d_in, const int* in_sizes, int n_in,
                              void* d_out, int out_size, void* d_ws, size_t ws_size,
                              hipStream_t stream) {
    (void)in_sizes; (void)n_in; (void)out_size; (void)ws_size;
    const float* x  = (const float*)d_in[0];   // [4][64][25][64][64]
    const float* Wm = (const float*)d_in[1];   // [192][64]
    float* out = (float*)d_out;
    float* ws  = (float*)d_ws;                 // ~70 KB used

    k_prep<<<dim3(1), dim3(256), 0, stream>>>(Wm, ws);
    k_gram_stats<<<dim3(25, 25, 4), dim3(128), 0, stream>>>(x, ws);
    k_softmax<<<dim3(4), dim3(32), 0, stream>>>(ws);
    k_out<<<dim3(512, 1, 4), dim3(256), 0, stream>>>(x, Wm, ws, out);
}